// GraphAttentionLayerv2_45277545234535
// MI455X (gfx1250) — compile-verified
//
#include <hip/hip_runtime.h>
#include <math.h>

// ---------------------------------------------------------------------------
// GAT layer for MI455X (gfx1250, wave32, WMMA).
//   h:[50000,256] f32, edge_index:[2,800000] i32, W:[256,64] f32, a:[128,1] f32
// Pipeline:
//   k_init      : zero h_prime/denom, set segment-max keys to enc(-inf)
//   k_gemm      : Wh = leaky_relu(h@W) via v_wmma_f32_16x16x4_f32 (+ e1,e2 fused)
//                 W staged in LDS in K-pair layout so B-frags are single b64 loads
//   k_edge_max  : e = e1[src]+e2[dst]; atomicMax per-src (ordered-uint trick)
//   k_edge_exp  : e_exp = exp(e - m[src]); atomicAdd denom[src]
//   k_aggregate : h_prime[src] += (e_exp/denom[src]) * Wh[dst] (f32 atomics, L2)
//   k_elu       : out = elu(h_prime)
// ---------------------------------------------------------------------------

typedef __attribute__((ext_vector_type(2))) float v2f;
typedef __attribute__((ext_vector_type(8))) float v8f;

#define N_NODES 50000
#define N_EDGES 800000
#define IN_F    256
#define OUT_F   64
#define LRELU_A 0.2f

// order-preserving float<->uint for atomicMax-based segment max
__device__ __forceinline__ unsigned fenc(float f) {
  unsigned u = __float_as_uint(f);
  return (u & 0x80000000u) ? ~u : (u | 0x80000000u);
}
__device__ __forceinline__ float fdec(unsigned k) {
  return __uint_as_float((k & 0x80000000u) ? (k & 0x7fffffffu) : ~k);
}
#define ENC_NEG_INF 0x007FFFFFu  // fenc(-inf)

// ---------------------------------------------------------------------------
__global__ __launch_bounds__(256) void k_init(float* __restrict__ hp,
                                              unsigned* __restrict__ mkey,
                                              float* __restrict__ denom) {
  int idx = blockIdx.x * blockDim.x + threadIdx.x;
  if (idx < N_NODES * OUT_F) hp[idx] = 0.0f;
  if (idx < N_NODES) { mkey[idx] = ENC_NEG_INF; denom[idx] = 0.0f; }
}

// ---------------------------------------------------------------------------
// One wave computes a 16x64 strip of Wh with 4 f32 WMMA accumulators.
// 4 waves / block. LDS holds W in K-pair-major form:
//   sWp[(k>>1)*OUT_F + n] = { W[k][n], W[k+1][n] }   (64 KB)
// so a B fragment (v2f = {B[kb][n], B[kb+1][n]}) is one aligned ds_load_b64.
__global__ __launch_bounds__(128) void k_gemm(const float* __restrict__ h,
                                              const float* __restrict__ W,
                                              const float* __restrict__ av,
                                              float* __restrict__ Wh,
                                              float* __restrict__ e1,
                                              float* __restrict__ e2) {
  __shared__ v2f sWp[(IN_F / 2) * OUT_F];            // 128*64 pairs = 64 KB
  for (int i = threadIdx.x; i < (IN_F / 2) * OUT_F; i += 128) {
    const int kp = i >> 6;                           // K-pair index
    const int n  = i & 63;
    v2f p;
    p.x = W[(2 * kp) * OUT_F + n];
    p.y = W[(2 * kp + 1) * OUT_F + n];
    sWp[i] = p;
  }
  __syncthreads();

  const int lane    = threadIdx.x & 31;
  const int wv      = threadIdx.x >> 5;
  const int rowBase = (blockIdx.x * 4 + wv) * 16;
  const bool valid  = rowBase < N_NODES;             // wave-uniform (N%16==0)
  const int ln      = lane & 15;
  const int hi      = lane >> 4;

  // A-fragment row for this lane (clamped for tail waves; stores are skipped)
  const float* hrow = h + (size_t)((valid ? rowBase : 0) + ln) * IN_F;

  v8f acc0 = {}, acc1 = {}, acc2 = {}, acc3 = {};

  for (int k0 = 0; k0 < IN_F; k0 += 4) {
    // A frag: lane holds {h[row][kb], h[row][kb+1]}, kb = k0 + 2*hi  -> b64 load
    const v2f af = *(const v2f*)(hrow + k0 + 2 * hi);

    // B frags: one ds_load_b64 each, no repacking needed
    const v2f* bp = &sWp[((k0 >> 1) + hi) * OUT_F + ln];
    const v2f bf0 = bp[0];
    const v2f bf1 = bp[16];
    const v2f bf2 = bp[32];
    const v2f bf3 = bp[48];

    acc0 = __builtin_amdgcn_wmma_f32_16x16x4_f32(false, af, false, bf0, (short)0, acc0, false, false);
    acc1 = __builtin_amdgcn_wmma_f32_16x16x4_f32(false, af, false, bf1, (short)0, acc1, false, false);
    acc2 = __builtin_amdgcn_wmma_f32_16x16x4_f32(false, af, false, bf2, (short)0, acc2, false, false);
    acc3 = __builtin_amdgcn_wmma_f32_16x16x4_f32(false, af, false, bf3, (short)0, acc3, false, false);
  }

  // LeakyReLU on the accumulators (reference applies it before e1/e2 and store)
#pragma unroll
  for (int r = 0; r < 8; ++r) {
    acc0[r] = acc0[r] > 0.f ? acc0[r] : LRELU_A * acc0[r];
    acc1[r] = acc1[r] > 0.f ? acc1[r] : LRELU_A * acc1[r];
    acc2[r] = acc2[r] > 0.f ? acc2[r] : LRELU_A * acc2[r];
    acc3[r] = acc3[r] > 0.f ? acc3[r] : LRELU_A * acc3[r];
  }

  if (!valid) return;                                // wave-uniform exit

  // attention projection vectors for this lane's N columns
  const float a10 = av[ln],       a11 = av[16 + ln],
              a12 = av[32 + ln],  a13 = av[48 + ln];
  const float a20 = av[64 + ln],  a21 = av[80 + ln],
              a22 = av[96 + ln],  a23 = av[112 + ln];

#pragma unroll
  for (int r = 0; r < 8; ++r) {
    const int rowM = rowBase + r + 8 * hi;           // C-layout: VGPR r -> M
    // store Wh strip: lane ln writes column ln of each 16-wide N tile
    float* wr = Wh + (size_t)rowM * OUT_F + ln;
    wr[0]  = acc0[r];
    wr[16] = acc1[r];
    wr[32] = acc2[r];
    wr[48] = acc3[r];

    // fused e1/e2 = Wh @ a halves: reduce over N (16 lanes in this half)
    float s1 = acc0[r] * a10 + acc1[r] * a11 + acc2[r] * a12 + acc3[r] * a13;
    float s2 = acc0[r] * a20 + acc1[r] * a21 + acc2[r] * a22 + acc3[r] * a23;
#pragma unroll
    for (int off = 8; off >= 1; off >>= 1) {
      s1 += __shfl_xor(s1, off, 16);
      s2 += __shfl_xor(s2, off, 16);
    }
    if (ln == 0) { e1[rowM] = s1; e2[rowM] = s2; }
  }
}

// ---------------------------------------------------------------------------
__global__ __launch_bounds__(256) void k_edge_max(const int* __restrict__ ei,
                                                  const float* __restrict__ e1,
                                                  const float* __restrict__ e2,
                                                  float* __restrict__ eb,
                                                  unsigned* __restrict__ mkey) {
  int e = blockIdx.x * blockDim.x + threadIdx.x;
  if (e >= N_EDGES) return;
  const int src = ei[e];
  const int dst = ei[N_EDGES + e];
  const float val = e1[src] + e2[dst];
  eb[e] = val;
  atomicMax(&mkey[src], fenc(val));
}

__global__ __launch_bounds__(256) void k_edge_exp(const int* __restrict__ ei,
                                                  const unsigned* __restrict__ mkey,
                                                  float* __restrict__ eb,
                                                  float* __restrict__ denom) {
  int e = blockIdx.x * blockDim.x + threadIdx.x;
  if (e >= N_EDGES) return;
  const int src = ei[e];
  const float ex = __expf(eb[e] - fdec(mkey[src]));
  eb[e] = ex;
  atomicAdd(&denom[src], ex);
}

// ---------------------------------------------------------------------------
// One wave32 per edge; each lane handles 2 consecutive features (64 total).
// Wh (12.8 MB) is L2-resident on MI455X, so gathers + f32 atomics stay on-chip.
__global__ __launch_bounds__(256) void k_aggregate(const int* __restrict__ ei,
                                                   const float* __restrict__ eb,
                                                   const float* __restrict__ denom,
                                                   const float* __restrict__ Wh,
                                                   float* __restrict__ hp) {
  const int wave = (blockIdx.x * blockDim.x + threadIdx.x) >> 5;
  const int lane = threadIdx.x & 31;
  if (wave >= N_EDGES) return;
  const int src = ei[wave];
  const int dst = ei[N_EDGES + wave];
  const float attn = eb[wave] / denom[src];

  const float2 w = ((const float2*)(Wh + (size_t)dst * OUT_F))[lane];
  float* d = hp + (size_t)src * OUT_F + 2 * lane;
  atomicAdd(&d[0], attn * w.x);
  atomicAdd(&d[1], attn * w.y);
}

// ---------------------------------------------------------------------------
__global__ __launch_bounds__(256) void k_elu(const float* __restrict__ hp,
                                             float* __restrict__ out) {
  int idx = blockIdx.x * blockDim.x + threadIdx.x;
  if (idx >= N_NODES * OUT_F) return;
  const float x = hp[idx];
  out[idx] = x > 0.f ? x : expm1f(x);
}

// ---------------------------------------------------------------------------
extern "C" void kernel_launch(void* const* d_in, const int* in_sizes, int n_in,
                              void* d_out, int out_size, void* d_ws, size_t ws_size,
                              hipStream_t stream) {
  const float* h  = (const float*)d_in[0];
  const int*   ei = (const int*)d_in[1];
  const float* W  = (const float*)d_in[2];
  const float* a  = (const float*)d_in[3];
  float* out = (float*)d_out;

  char* ws = (char*)d_ws;
  float*    Wh    = (float*)(ws);                                 // 12,800,000 B
  float*    e1    = (float*)(ws + 12800000);                      //    200,000 B
  float*    e2    = (float*)(ws + 13000000);                      //    200,000 B
  unsigned* mkey  = (unsigned*)(ws + 13200000);                   //    200,000 B
  float*    denom = (float*)(ws + 13400000);                      //    200,000 B
  float*    eb    = (float*)(ws + 13600000);                      //  3,200,000 B
  float*    hp    = (float*)(ws + 16800000);                      // 12,800,000 B
  // total workspace: 29,600,000 bytes

  const int nodeElems = N_NODES * OUT_F;

  k_init<<<(nodeElems + 255) / 256, 256, 0, stream>>>(hp, mkey, denom);

  const int nTiles = N_NODES / 16;                   // 3125 (N divisible by 16)
  k_gemm<<<(nTiles + 3) / 4, 128, 0, stream>>>(h, W, a, Wh, e1, e2);

  k_edge_max<<<(N_EDGES + 255) / 256, 256, 0, stream>>>(ei, e1, e2, eb, mkey);
  k_edge_exp<<<(N_EDGES + 255) / 256, 256, 0, stream>>>(ei, mkey, eb, denom);

  // 8 edges (waves) per 256-thread block
  k_aggregate<<<N_EDGES / 8, 256, 0, stream>>>(ei, eb, denom, Wh, hp);

  k_elu<<<(nodeElems + 255) / 256, 256, 0, stream>>>(hp, out);
}